// DiagonalVariational_86981677678680
// MI455X (gfx1250) — compile-verified
//
#include <hip/hip_runtime.h>

// out[n,d] = m[d] + sqrt(L[d]^2 + 1e-6) * eps[n,d]
// D = 16384, N = 2048, all f32. Pure streaming: ~268 MB HBM traffic,
// roofline floor ~11.5 us @ 23.3 TB/s. Memory-bound; no matrix structure
// (no K-reduction), so WMMA is inapplicable -- the win is the memory path:
// coalesced B128 NT loads/stores, v_pk_fma_f32, hoisted v_sqrt_f32.

#define D_DIM    16384
#define N_ROWS   2048
#define JITTER_F 1e-6f

typedef float f4 __attribute__((ext_vector_type(4)));

#define THREADS   256                       // 8 wave32s per block
#define COLS_PER_BLOCK (THREADS * 4)        // 1024 columns per block
#define GRID_X    (D_DIM / COLS_PER_BLOCK)  // 16
#define GRID_Y    128
#define ROWS_PER_BLOCK (N_ROWS / GRID_Y)    // 16 contiguous rows per block

__global__ __launch_bounds__(THREADS)
void DiagonalVariational_kernel(const float* __restrict__ m,
                                const float* __restrict__ L,
                                const float* __restrict__ eps,
                                float* __restrict__ out)
{
    // Each thread owns 4 consecutive columns d0..d0+3 for a strip of rows.
    const int d0 = (blockIdx.x * THREADS + threadIdx.x) * 4;

    // Per-column constants: loaded once, reused across all rows in the strip.
    // m/L are tiny (64 KiB each) and shared by all blocks -> regular (RT) loads.
    const f4 mv = *reinterpret_cast<const f4*>(m + d0);
    const f4 lv = *reinterpret_cast<const f4*>(L + d0);

    // Hoisted scale: raw v_sqrt_f32 (~1 ulp). Argument >= 1e-6, far from the
    // denorm range, so the IEEE fixup expansion of sqrtf() is dead code here.
    f4 sv;
    sv.x = __builtin_amdgcn_sqrtf(__builtin_fmaf(lv.x, lv.x, JITTER_F));
    sv.y = __builtin_amdgcn_sqrtf(__builtin_fmaf(lv.y, lv.y, JITTER_F));
    sv.z = __builtin_amdgcn_sqrtf(__builtin_fmaf(lv.z, lv.z, JITTER_F));
    sv.w = __builtin_amdgcn_sqrtf(__builtin_fmaf(lv.w, lv.w, JITTER_F));

    const int row0 = blockIdx.y * ROWS_PER_BLOCK;

    const float* __restrict__ ep = eps + (size_t)row0 * D_DIM + d0;
    float* __restrict__       op = out + (size_t)row0 * D_DIM + d0;

    #pragma unroll 8
    for (int r = 0; r < ROWS_PER_BLOCK; ++r) {
        const f4* esrc = reinterpret_cast<const f4*>(ep + (size_t)r * D_DIM);
        f4*       dst  = reinterpret_cast<f4*>(op + (size_t)r * D_DIM);

        // Prefetch next row of the eps stream (gfx1250 global_prefetch_b8).
        if (r + 1 < ROWS_PER_BLOCK) {
            __builtin_prefetch(ep + (size_t)(r + 1) * D_DIM, 0, 0);
        }

        // Non-temporal: 256 MB stream can't live in 192 MB L2; don't evict
        // the hot m/L lines every block rereads.
        f4 e = __builtin_nontemporal_load(esrc);

        f4 res;   // lowers to 2x v_pk_fma_f32 per float4
        res.x = __builtin_fmaf(sv.x, e.x, mv.x);
        res.y = __builtin_fmaf(sv.y, e.y, mv.y);
        res.z = __builtin_fmaf(sv.z, e.z, mv.z);
        res.w = __builtin_fmaf(sv.w, e.w, mv.w);

        __builtin_nontemporal_store(res, dst);
    }
}

extern "C" void kernel_launch(void* const* d_in, const int* in_sizes, int n_in,
                              void* d_out, int out_size, void* d_ws, size_t ws_size,
                              hipStream_t stream) {
    const float* m   = (const float*)d_in[0];   // [D]
    const float* L   = (const float*)d_in[1];   // [D]  (log_diag_L)
    const float* eps = (const float*)d_in[2];   // [N, D]
    float* out       = (float*)d_out;           // [N, D]

    dim3 grid(GRID_X, GRID_Y);   // 16 x 128 = 2048 blocks
    dim3 block(THREADS);         // 256 threads = 8 wave32s
    DiagonalVariational_kernel<<<grid, block, 0, stream>>>(m, L, eps, out);
}